// S4Kernel_52905407152477
// MI455X (gfx1250) — compile-verified
//
#include <hip/hip_runtime.h>
#include <hip/hip_bf16.h>

typedef __attribute__((ext_vector_type(2))) float v2f;
typedef __attribute__((ext_vector_type(8))) float v8f;

#define BATCH   2
#define SEQLEN  2048
#define DMODEL  512
#define DSTATE  64
#define QC      16                 // chunk length == WMMA tile
#define NCHUNK  (SEQLEN / QC)      // 128
#define COLS    (BATCH * NCHUNK)   // 256 chunk-columns per channel
#define LT      (BATCH * SEQLEN)   // 4096

#if defined(__HIP_DEVICE_COMPILE__) && __has_builtin(__builtin_amdgcn_global_load_async_to_lds_b32)
#define HAVE_ASYNC_LDS 1
#endif

// direct global -> LDS copy (ASYNCcnt path) with scalar fallback
static __device__ __forceinline__ void async_ld_f32(const float* g, float* l) {
#if defined(HAVE_ASYNC_LDS)
    __builtin_amdgcn_global_load_async_to_lds_b32(
        (__attribute__((address_space(1))) int*)(int*)(float*)g,
        (__attribute__((address_space(3))) int*)(int*)l, 0, 0);
#else
    *l = *g;
#endif
}

static __device__ __forceinline__ void wait_async_lds() {
#if defined(HAVE_ASYNC_LDS)
#if __has_builtin(__builtin_amdgcn_s_wait_asynccnt)
    __builtin_amdgcn_s_wait_asynccnt(0);
#else
    asm volatile("s_wait_asynccnt 0x0" ::: "memory");
#endif
#endif
}

static __device__ __forceinline__ v8f wmma_f32x4(v2f a, v2f b, v8f c) {
    // V_WMMA_F32_16X16X4_F32 : D = A(16x4) * B(4x16) + C(16x16)
    return __builtin_amdgcn_wmma_f32_16x16x4_f32(
        false, a, false, b, (short)0, c, false, false);
}

// -------------------- 1) transpose u[4096,512] -> uT[512,4096] --------------
__global__ __launch_bounds__(256) void k_transpose_in(const float* __restrict__ u,
                                                      float* __restrict__ uT) {
    __shared__ float t[32][33];
    const int hb = blockIdx.x * 32;   // channel tile
    const int lb = blockIdx.y * 32;   // time tile
    #pragma unroll
    for (int i = 0; i < 4; ++i) {
        int l = lb + threadIdx.y + i * 8;
        int h = hb + threadIdx.x;
        async_ld_f32(&u[(size_t)l * DMODEL + h], &t[threadIdx.y + i * 8][threadIdx.x]);
    }
    wait_async_lds();
    __syncthreads();
    #pragma unroll
    for (int i = 0; i < 4; ++i) {
        int h = hb + threadIdx.y + i * 8;
        int l = lb + threadIdx.x;
        uT[(size_t)h * LT + l] = t[threadIdx.x][threadIdx.y + i * 8];
    }
}

// -------------------- 2) per-channel parameter matrices ---------------------
// M  [H][16][16] : causal Toeplitz kernel  M[t][s] = K[t-s]
// CA [H][16][64] : carry matrix           CA[t][n] = C[n] * A_d^(t+1)
// W  [H][64][16] : chunk input matrix     W[n][s]  = A_d^(15-s) * B_d[n]
// AP [H][64]     : A_d^16
__global__ __launch_bounds__(64) void k_precompute(const float* __restrict__ B_re,
                                                   const float* __restrict__ C_re,
                                                   const float* __restrict__ log_dt,
                                                   float* __restrict__ Mm,
                                                   float* __restrict__ CAm,
                                                   float* __restrict__ Wm,
                                                   float* __restrict__ APm) {
    const int h = blockIdx.x;
    const int n = threadIdx.x;
    const float A     = expf(-0.5f * log1pf((float)(n + 1) / (float)DSTATE));
    const float delta = expf(log_dt[h]);
    const float dA    = delta * A;
    const float A_d   = expf(dA);
    const float Bd    = (A_d - 1.0f) * B_re[n] * delta;
    const float Cn    = C_re[n];

    APm[h * DSTATE + n] = expf(16.0f * dA);
    #pragma unroll
    for (int s = 0; s < QC; ++s)
        Wm[((size_t)h * DSTATE + n) * QC + s] = expf((float)(15 - s) * dA) * Bd;
    #pragma unroll
    for (int t = 0; t < QC; ++t)
        CAm[((size_t)h * QC + t) * DSTATE + n] = Cn * expf((float)(t + 1) * dA);

    __shared__ float contrib[QC][DSTATE];
    __shared__ float Ks[QC];
    #pragma unroll
    for (int k = 0; k < QC; ++k)
        contrib[k][n] = Cn * expf((float)k * dA) * Bd;
    __syncthreads();
    if (n < QC) {
        float s = 0.0f;
        for (int j = 0; j < DSTATE; ++j) s += contrib[n][j];
        Ks[n] = s;
    }
    __syncthreads();
    #pragma unroll
    for (int q = 0; q < 4; ++q) {
        int idx = n * 4 + q;
        int t = idx >> 4, s2 = idx & 15;
        Mm[(size_t)h * 256 + idx] = (s2 <= t) ? Ks[t - s2] : 0.0f;
    }
}

// -------------------- 3) Phase A GEMM: Wc[h] = W[64x16] @ U[16x256] ---------
__global__ __launch_bounds__(128) void k_phaseA(const float* __restrict__ uT,
                                                const float* __restrict__ Wm,
                                                float* __restrict__ Wc) {
    const int ct = blockIdx.x;   // column tile (16 chunk-columns)
    const int h  = blockIdx.y;
    __shared__ float lu[16][17]; // lu[j][s] = U[s][col=ct*16+j], padded
    const int tid = threadIdx.x;
    const float* gsrc = uT + (size_t)h * LT + (size_t)ct * 256;
    #pragma unroll
    for (int f = tid; f < 256; f += 128)
        async_ld_f32(&gsrc[f], &lu[f >> 4][f & 15]);
    wait_async_lds();
    __syncthreads();

    const int lane    = tid & 31;
    const int wv      = tid >> 5;           // 4 waves: n-row tiles
    const int rowBase = wv * 16;
    const int mrow    = rowBase + (lane & 15);
    const int khalf   = (lane >> 4) << 1;   // lanes 16-31 carry K+2,K+3
    const int j       = lane & 15;
    const float* Wrow = Wm + ((size_t)h * DSTATE + mrow) * QC;

    v8f acc = {};
    #pragma unroll
    for (int kk = 0; kk < 4; ++kk) {
        int k0 = kk * 4 + khalf;
        v2f a; a.x = Wrow[k0];      a.y = Wrow[k0 + 1];
        v2f b; b.x = lu[j][k0];     b.y = lu[j][k0 + 1];
        acc = wmma_f32x4(a, b, acc);
    }
    const int mhalf = (lane >> 4) * 8;
    #pragma unroll
    for (int r = 0; r < 8; ++r) {
        int m = r + mhalf;
        Wc[((size_t)h * DSTATE + rowBase + m) * COLS + ct * 16 + j] = acc[r];
    }
}

// -------------------- 4) cross-chunk scan (in place: w_c -> x_start[c]) -----
__global__ __launch_bounds__(256) void k_scan(const float* __restrict__ APm,
                                              float* __restrict__ Wc) {
    const int idx = blockIdx.x * blockDim.x + threadIdx.x; // h*64+n
    const float a = APm[idx];
    float* row = Wc + (size_t)idx * COLS;
    #pragma unroll
    for (int b = 0; b < BATCH; ++b) {
        __builtin_prefetch(row + b * NCHUNK, 0, 0);  // global_prefetch_b8
        float x = 0.0f;
        for (int c = 0; c < NCHUNK; ++c) {
            float w = row[b * NCHUNK + c];
            row[b * NCHUNK + c] = x;        // state entering chunk c
            x = fmaf(a, x, w);              // A^16 * x + w_c
        }
    }
}

// -------------------- 5) Phase C: Y = M@U + CA@X + D*u ----------------------
__global__ __launch_bounds__(32) void k_phaseC(const float* __restrict__ uT,
                                               const float* __restrict__ Mm,
                                               const float* __restrict__ CAm,
                                               const float* __restrict__ Wc,
                                               const float* __restrict__ Dv,
                                               float* __restrict__ yT) {
    const int ct   = blockIdx.x;
    const int h    = blockIdx.y;
    const int lane = threadIdx.x;
    const int trow  = lane & 15;
    const int khalf = (lane >> 4) << 1;
    const int j     = lane & 15;

    const float* Mrow  = Mm  + (size_t)h * 256 + trow * QC;
    const float* CArow = CAm + ((size_t)h * QC + trow) * DSTATE;
    const float* ub    = uT  + (size_t)h * LT + (size_t)ct * 256;  // ub[j*16+s]
    const float* Xb    = Wc  + (size_t)h * DSTATE * COLS + ct * 16;

    v8f acc = {};
    #pragma unroll
    for (int kk = 0; kk < 4; ++kk) {        // intra-chunk causal Toeplitz
        int k0 = kk * 4 + khalf;
        v2f a; a.x = Mrow[k0];          a.y = Mrow[k0 + 1];
        v2f b; b.x = ub[j * 16 + k0];   b.y = ub[j * 16 + k0 + 1];
        acc = wmma_f32x4(a, b, acc);
    }
    #pragma unroll
    for (int kk = 0; kk < 16; ++kk) {       // carry from chunk-start state
        int k0 = kk * 4 + khalf;
        v2f a; a.x = CArow[k0];               a.y = CArow[k0 + 1];
        v2f b; b.x = Xb[(size_t)k0 * COLS + j];
               b.y = Xb[(size_t)(k0 + 1) * COLS + j];
        acc = wmma_f32x4(a, b, acc);
    }
    const float Dh = Dv[h];
    const int mhalf = (lane >> 4) * 8;
    #pragma unroll
    for (int r = 0; r < 8; ++r) {
        int t = r + mhalf;
        float uval = ub[j * 16 + t];
        yT[(size_t)h * LT + (size_t)ct * 256 + j * 16 + t] = acc[r] + Dh * uval;
    }
}

// -------------------- 6) transpose yT[512,4096] -> y[4096,512] --------------
__global__ __launch_bounds__(256) void k_transpose_out(const float* __restrict__ yT,
                                                       float* __restrict__ y) {
    __shared__ float t[32][33];
    const int hb = blockIdx.x * 32;
    const int lb = blockIdx.y * 32;
    #pragma unroll
    for (int i = 0; i < 4; ++i) {
        int h = hb + threadIdx.y + i * 8;
        int l = lb + threadIdx.x;
        async_ld_f32(&yT[(size_t)h * LT + l], &t[threadIdx.y + i * 8][threadIdx.x]);
    }
    wait_async_lds();
    __syncthreads();
    #pragma unroll
    for (int i = 0; i < 4; ++i) {
        int l = lb + threadIdx.y + i * 8;
        int h = hb + threadIdx.x;
        y[(size_t)l * DMODEL + h] = t[threadIdx.x][threadIdx.y + i * 8];
    }
}

extern "C" void kernel_launch(void* const* d_in, const int* in_sizes, int n_in,
                              void* d_out, int out_size, void* d_ws, size_t ws_size,
                              hipStream_t stream) {
    const float* u      = (const float*)d_in[0];
    const float* B_re   = (const float*)d_in[1];
    const float* C_re   = (const float*)d_in[2];
    const float* log_dt = (const float*)d_in[3];
    const float* Dv     = (const float*)d_in[4];
    float* y = (float*)d_out;

    float* ws  = (float*)d_ws;
    float* uT  = ws;                                  // 512*4096
    float* yT  = uT  + (size_t)DMODEL * LT;           // 512*4096
    float* Mm  = yT  + (size_t)DMODEL * LT;           // 512*256
    float* CAm = Mm  + (size_t)DMODEL * 256;          // 512*1024
    float* Wm  = CAm + (size_t)DMODEL * QC * DSTATE;  // 512*1024
    float* APm = Wm  + (size_t)DMODEL * DSTATE * QC;  // 512*64
    float* Wc  = APm + (size_t)DMODEL * DSTATE;       // 512*64*256

    k_transpose_in <<<dim3(DMODEL / 32, LT / 32), dim3(32, 8), 0, stream>>>(u, uT);
    k_precompute   <<<DMODEL, DSTATE, 0, stream>>>(B_re, C_re, log_dt, Mm, CAm, Wm, APm);
    k_phaseA       <<<dim3(COLS / 16, DMODEL), 128, 0, stream>>>(uT, Wm, Wc);
    k_scan         <<<(DMODEL * DSTATE) / 256, 256, 0, stream>>>(APm, Wc);
    k_phaseC       <<<dim3(COLS / 16, DMODEL), 32, 0, stream>>>(uT, Mm, CAm, Wc, Dv, yT);
    k_transpose_out<<<dim3(DMODEL / 32, LT / 32), dim3(32, 8), 0, stream>>>(yT, y);
}